// PManifoldLayer_66477503807693
// MI455X (gfx1250) — compile-verified
//
#include <hip/hip_runtime.h>
#include <math.h>

// Problem constants (reference: B,N,M,K,H = 32,2048,16,32,2)
#define B_  32
#define N_  2048
#define M_  16
#define K_  32
#define H_  2
#define ROWW 17           // dgm row width = 1 (hom) + M (coords)
#define EPS 1e-7f

#define CH_ROWS 128                       // n-rows per TDM chunk
#define CH_ELEMS (CH_ROWS * ROWW)         // 2176 f32 elements
#define CH_BYTES (CH_ELEMS * 4)           // 8704 B per buffer
#define N_CHUNKS (N_ / CH_ROWS)           // 16

typedef __attribute__((ext_vector_type(2))) float v2f;
typedef __attribute__((ext_vector_type(8))) float v8f;

#if defined(__has_builtin)
#if __has_builtin(__builtin_amdgcn_tensor_load_to_lds) && \
    __has_builtin(__builtin_amdgcn_s_wait_tensorcnt)
#define USE_TDM 1
#endif
#endif

#ifdef USE_TDM
typedef unsigned tdm_v4u __attribute__((ext_vector_type(4)));
typedef int      tdm_v8i __attribute__((ext_vector_type(8)));
typedef int      tdm_v4i __attribute__((ext_vector_type(4)));

// Issue a TDM load of a 1D run of `tile_elems` f32 elements starting at gsrc
// into LDS byte offset lds_off. D# built per CDNA5 ISA Ch.8:
//  group0: [1:0]=count=1 | [63:32]=lds_addr | [120:64]=global_addr | [127:126]=type=2
//  group1: [17:16]=data_size(2 -> 4B) | tensor_dim0 @ [79:48] | tensor_dim1 @ [111:80]
//          tile_dim0 @ [127:112] | tensor_dim0_stride @ [207:160]
__device__ __forceinline__ void tdm_load_1d(const float* gsrc, unsigned lds_off,
                                            int tile_elems) {
    unsigned long long ga = (unsigned long long)(uintptr_t)gsrc;
    tdm_v4u g0 = { 1u,
                   lds_off,
                   (unsigned)(ga & 0xffffffffull),
                   (unsigned)((ga >> 32) & 0x01ffffffull) | 0x80000000u };
    const unsigned tdim0 = (unsigned)(N_ * ROWW);          // elements in tensor
    tdm_v8i g1 = { (int)0x00020000u,                        // data_size = 4B
                   (int)((tdim0 & 0xffffu) << 16),          // tensor_dim0[15:0]
                   (int)(((tdim0 >> 16) & 0xffffu) | (1u << 16)), // td0 hi | td1=1
                   (int)(((unsigned)tile_elems & 0xffffu) << 16), // tile_dim0
                   0,                                       // tile_dim1/2 unused
                   (int)tdim0,                              // tensor_dim0_stride lo
                   0, 0 };
    tdm_v4i gz4 = { 0, 0, 0, 0 };
    tdm_v8i gz8 = { 0, 0, 0, 0, 0, 0, 0, 0 };
    __builtin_amdgcn_tensor_load_to_lds(g0, g1, gz4, gz4, gz8, 0);
}
#endif

// ---------------------------------------------------------------------------
// Kernel 0: per-batch first all-zero row index  ->  valid = (n < n0[b])
// ---------------------------------------------------------------------------
__global__ void pm_firstzero(const float* __restrict__ dgm, int* __restrict__ n0) {
    int b = blockIdx.x;
    __shared__ int smin;
    if (threadIdx.x == 0) smin = N_;
    __syncthreads();
    for (int n = threadIdx.x; n < N_; n += blockDim.x) {
        const float* row = dgm + ((size_t)b * N_ + n) * ROWW;
        bool anynz = false;
#pragma unroll
        for (int j = 0; j < ROWW; ++j) anynz |= (row[j] != 0.0f);
        if (!anynz) atomicMin(&smin, n);
    }
    __syncthreads();
    if (threadIdx.x == 0) n0[b] = smin;
}

// ---------------------------------------------------------------------------
// Kernel 1 (WMMA GEMM #1, fused gain):
//   zn2[b,n,k] = sum_m y^2 * theta^2      via v_wmma_f32_16x16x4_f32
//   a[b,k,n]   = w[b,n] * g(zn2)
// ---------------------------------------------------------------------------
__global__ void pm_zn2_gain(const float* __restrict__ dgm,
                            const float* __restrict__ theta,
                            const float* __restrict__ class_w,
                            const int*   __restrict__ n0,
                            float*       __restrict__ aout) {
    __shared__ float th2[K_ * M_];
    __shared__ float wsm[128];
    const int b    = blockIdx.y;
    const int nblk = blockIdx.x * 128;
    const int t    = threadIdx.x;

    for (int i = t; i < K_ * M_; i += 256) { float v = theta[i]; th2[i] = v * v; }

    const int nz = n0[b];
    const float cw0 = class_w[0], cw1 = class_w[1];
    if (t < 128) {
        int n = nblk + t;
        float h = dgm[((size_t)b * N_ + n) * ROWW];
        int hi = (int)h; hi = hi < 0 ? 0 : (hi > H_ - 1 ? H_ - 1 : hi);
        float w = (hi == 0) ? cw0 : cw1;
        wsm[t] = (n < nz) ? w : 0.0f;
    }
    __syncthreads();

    const int wave  = t >> 5;
    const int lane  = t & 31;
    const int ntile = nblk + wave * 16;
    const int idx   = lane & 15;   // A row (n) AND B col (k)
    const int hi16  = lane >> 4;
    const int kb    = 2 * hi16;    // lane's K-pair base

    const float* yrow = dgm + ((size_t)b * N_ + ntile + idx) * ROWW + 1;
    float y2[M_];
#pragma unroll
    for (int m = 0; m < M_; ++m) { float v = yrow[m]; y2[m] = v * v; }

    for (int k0 = 0; k0 < K_; k0 += 16) {
        v8f c = {};
#pragma unroll
        for (int mc = 0; mc < M_; mc += 4) {
            v2f A  = { y2[mc + kb], y2[mc + kb + 1] };
            v2f Bm = { th2[(k0 + idx) * M_ + mc + kb],
                       th2[(k0 + idx) * M_ + mc + kb + 1] };
            c = __builtin_amdgcn_wmma_f32_16x16x4_f32(false, A, false, Bm,
                                                      (short)0, c, false, false);
        }
#pragma unroll
        for (int r = 0; r < 8; ++r) {
            int   n   = ntile + r + 8 * hi16;
            float t2  = c[r];
            float s   = sqrtf(1.0f + t2);
            float den = 1.0f + s;
            float rr  = sqrtf(t2) / den;                        // xn
            float rc  = fminf(rr, 1.0f - EPS);
            float at  = 0.5f * logf((1.0f + rc) / (1.0f - rc)); // artanh
            float g   = at / fmaxf(rr, EPS) / den;
            aout[((size_t)b * K_ + (k0 + idx)) * N_ + n] = wsm[n - nblk] * g;
        }
    }
}

// ---------------------------------------------------------------------------
// Kernel 2 (WMMA GEMM #2, contraction N=2048):
//   P[b,k,m] = sum_n a[b,k,n] * y[b,n,m]
// TDM path: double-buffered tensor_load_to_lds of 128-row y panels; wave 0
// drives the DMA (TENSORcnt), __syncthreads publishes to wave 1; B operand
// from LDS, A operand contiguous f32x2 per lane from global.
// ---------------------------------------------------------------------------
__global__ void pm_contract(const float* __restrict__ dgm,
                            const float* __restrict__ aarr,
                            float*       __restrict__ P) {
    const int b    = blockIdx.x;
    const int wave = threadIdx.x >> 5;
    const int lane = threadIdx.x & 31;
    const int k0   = wave * 16;
    const int idx  = lane & 15;    // A row (k) AND B col (m)
    const int hi16 = lane >> 4;
    const int kb   = 2 * hi16;

    const float* arow = aarr + ((size_t)b * K_ + k0 + idx) * N_;
    v8f c = {};

#ifdef USE_TDM
    extern __shared__ float ybuf[];                 // 2 x CH_ELEMS floats
    const float* gbase = dgm + (size_t)b * N_ * ROWW;

    if (wave == 0) tdm_load_1d(gbase, 0u, CH_ELEMS);            // chunk 0 -> buf 0

    for (int ch = 0; ch < N_CHUNKS; ++ch) {
        if (wave == 0) {
            if (ch + 1 < N_CHUNKS) {
                tdm_load_1d(gbase + (size_t)(ch + 1) * CH_ELEMS,
                            (unsigned)(((ch + 1) & 1) * CH_BYTES), CH_ELEMS);
                __builtin_amdgcn_s_wait_tensorcnt(1);   // chunk `ch` done
            } else {
                __builtin_amdgcn_s_wait_tensorcnt(0);   // last chunk done
            }
        }
        __syncthreads();                            // chunk `ch` visible to all
        const float* yl = ybuf + (ch & 1) * CH_ELEMS + 1;  // skip hom column
        const int nbase = ch * CH_ROWS;
#pragma unroll 4
        for (int nl = 0; nl < CH_ROWS; nl += 4) {
            __builtin_prefetch(arow + nbase + nl + 256, 0, 1); // global_prefetch_b8
            v2f A  = { arow[nbase + nl + kb], arow[nbase + nl + kb + 1] };
            v2f Bm = { yl[(nl + kb) * ROWW + idx],
                       yl[(nl + kb + 1) * ROWW + idx] };
            c = __builtin_amdgcn_wmma_f32_16x16x4_f32(false, A, false, Bm,
                                                      (short)0, c, false, false);
        }
        __syncthreads();                            // buffer reuse safe
    }
#else
    const float* ybase = dgm + (size_t)b * N_ * ROWW + 1;
    for (int nc = 0; nc < N_; nc += 4) {
        __builtin_prefetch(arow + nc + 256, 0, 1);
        v2f A  = { arow[nc + kb], arow[nc + kb + 1] };
        v2f Bm = { ybase[(size_t)(nc + kb)     * ROWW + idx],
                   ybase[(size_t)(nc + kb + 1) * ROWW + idx] };
        c = __builtin_amdgcn_wmma_f32_16x16x4_f32(false, A, false, Bm,
                                                  (short)0, c, false, false);
    }
#endif

#pragma unroll
    for (int r = 0; r < 8; ++r) {
        int k = k0 + r + 8 * hi16;
        P[((size_t)b * K_ + k) * M_ + idx] = c[r];
    }
}

// ---------------------------------------------------------------------------
// Kernel 3: T = theta .* P ; serial cumsum over rows r=b*K+k ; tanh/Poincare.
// ---------------------------------------------------------------------------
__global__ void pm_final(const float* __restrict__ theta,
                         const float* __restrict__ P,
                         float*       __restrict__ out) {
    __shared__ float S[B_ * K_ * M_];     // 1024 x 16 floats = 64 KB
    const int t = threadIdx.x;
    if (t < M_) {
        float acc = 0.0f;
        for (int r = 0; r < B_ * K_; ++r) {
            int k = r & (K_ - 1);
            acc += theta[k * M_ + t] * P[r * M_ + t];
            S[r * M_ + t] = acc;
        }
    }
    __syncthreads();
    {
        const int r = t;
        float s[M_]; float sn2 = 0.0f;
#pragma unroll
        for (int m = 0; m < M_; ++m) { s[m] = S[r * M_ + m]; sn2 += s[m] * s[m]; }
        float sn = sqrtf(sn2);
        float sc = tanhf(sn) / fmaxf(sn, EPS);
        float x[M_]; float xn2 = 0.0f;
#pragma unroll
        for (int m = 0; m < M_; ++m) { x[m] = sc * s[m]; xn2 += x[m] * x[m]; }
        float d = 2.0f / (1.0f - xn2);
#pragma unroll
        for (int m = 0; m < M_; ++m) out[r * M_ + m] = d * x[m];
    }
}

// ---------------------------------------------------------------------------
extern "C" void kernel_launch(void* const* d_in, const int* in_sizes, int n_in,
                              void* d_out, int out_size, void* d_ws, size_t ws_size,
                              hipStream_t stream) {
    const float* dgm     = (const float*)d_in[0];
    const float* theta   = (const float*)d_in[1];
    const float* class_w = (const float*)d_in[2];
    float* out = (float*)d_out;

    char* ws = (char*)d_ws;
    size_t aBytes = (size_t)B_ * K_ * N_ * sizeof(float);   // 8 MB
    size_t pBytes = (size_t)B_ * K_ * M_ * sizeof(float);   // 64 KB
    float* a  = (float*)ws;
    float* P  = (float*)(ws + aBytes);
    int*   n0 = (int*)(ws + aBytes + pBytes);

    pm_firstzero<<<B_, 256, 0, stream>>>(dgm, n0);
    dim3 g1(N_ / 128, B_);
    pm_zn2_gain<<<g1, 256, 0, stream>>>(dgm, theta, class_w, n0, a);
    pm_contract<<<B_, 64, 2 * CH_BYTES, stream>>>(dgm, a, P);
    pm_final<<<1, B_ * K_, 0, stream>>>(theta, P, out);
}